// DeepseekV4Attention_4123168604685
// MI455X (gfx1250) — compile-verified
//
#include <hip/hip_runtime.h>
#include <math.h>

// ---------------- problem constants (from reference) ----------------
#define Bc    2
#define Sc    2048
#define Hc    2048
#define NHc   16
#define NKVc  4
#define HDc   128
#define NOPEc 64
#define QLRc  1024
#define OLRc  1024
#define OGc   8
#define KVP   2064            // S+1 = 2049 padded to 16*129 (max tiles needed)
#define SCALEc 0.08838834764831844f   // 128^-0.5
#define EPSc  1e-6f

typedef __attribute__((ext_vector_type(2))) float v2f;
typedef __attribute__((ext_vector_type(8))) float v8f;
typedef unsigned int v4u __attribute__((ext_vector_type(4)));
typedef int v4i __attribute__((ext_vector_type(4)));
typedef int v8i __attribute__((ext_vector_type(8)));

__device__ __forceinline__ v8f wmma_f32(v2f a, v2f b, v8f c) {
  // (neg_a, A, neg_b, B, c_mod, C, reuse_a, reuse_b) -> v_wmma_f32_16x16x4_f32
  return __builtin_amdgcn_wmma_f32_16x16x4_f32(false, a, false, b, (short)0, c, false, false);
}

// ---------------------------------------------------------------------
// Tensor Data Mover: 2-D fp32 tile load Global -> LDS with LDS padding.
// D# packing per CDNA5 ISA.
//   pad_interval_code: pad after 2^(code+1) dwords (4 -> 32, 6 -> 128)
//   pad_amount_code:   pad of (code+1) dwords      (3 -> 4 dwords)
// Issue once per wave (EXEC ignored); tracked with TENSORcnt.
// This toolchain exposes the 6-arg builtin: (g0, g1, g2, g3, g4, cpol).
// ---------------------------------------------------------------------
__device__ __forceinline__ void tdm_load_2d_f32(unsigned lds_addr, const float* gptr,
                                                unsigned tile_w, unsigned tile_h,
                                                unsigned row_stride_elems,
                                                unsigned pad_interval_code,
                                                unsigned pad_amount_code)
{
  unsigned long long ga = (unsigned long long)(uintptr_t)gptr;
  v4u g0;
  g0.x = 1u;                                    // count=1 (valid), user mode
  g0.y = lds_addr;                              // D#.lds_addr (bytes)
  g0.z = (unsigned)ga;                          // global_addr[31:0]  (bits 95:64)
  g0.w = (unsigned)((ga >> 32) & 0x01FFFFFFu)   // global_addr[56:32] (bits 120:96)
       | (2u << 30);                            // type = 2 ("image")
  v8i g1;
  g1[0] = (int)((2u << 16)                      // data_size = 4 bytes
              | (1u << 20)                      // pad_enable (LDS dest padding)
              | (pad_interval_code << 22)
              | (pad_amount_code << 25));
  g1[1] = (int)((tile_w & 0xFFFFu) << 16);      // tensor_dim0[15:0] @ bits 63:48
  g1[2] = (int)(((tile_w >> 16) & 0xFFFFu)      // tensor_dim0[31:16]
              | ((tile_h & 0xFFFFu) << 16));    // tensor_dim1[15:0]
  g1[3] = (int)(((tile_h >> 16) & 0xFFFFu)      // tensor_dim1[31:16]
              | ((tile_w & 0xFFFFu) << 16));    // tile_dim0 @ bits 127:112
  g1[4] = (int)(tile_h & 0xFFFFu);              // tile_dim1 (tile_dim2 = 0)
  g1[5] = (int)row_stride_elems;                // tensor_dim0_stride[31:0]
  g1[6] = 0;                                    // stride0 hi / dim1_stride lo
  g1[7] = 0;                                    // dim1_stride hi
  v4i z4 = {0, 0, 0, 0};
  v8i z8 = {0, 0, 0, 0, 0, 0, 0, 0};
  __builtin_amdgcn_tensor_load_to_lds(g0, g1, z4, z4, z8, 0);
}

// =====================================================================
// Generic NT GEMM: C[M,N] = A[M,K] * B[N,K]^T   (all row-major, fp32)
// block = 256 threads (8 waves), workgroup tile 128(M) x 64(N), KT = 32.
// TDM double-buffered LDS staging; each wave computes 32x32 via 2x2 WMMA.
// Requires M%128==0, N%64==0, K%32==0 (true for all calls here).
// =====================================================================
__global__ __launch_bounds__(256)
void gemm_nt_wmma(const float* __restrict__ A, const float* __restrict__ B,
                  float* __restrict__ C,
                  int M, int N, int K, int lda, int ldb, int ldc,
                  long long aBatch, long long bBatch, long long cBatch)
{
  A += (long long)blockIdx.z * aBatch;
  B += (long long)blockIdx.z * bBatch;
  C += (long long)blockIdx.z * cBatch;

  __shared__ float As[2][128][36];   // 32 cols + 4 pad (TDM pad codes 4,3)
  __shared__ float Bs[2][64][36];

  const int tid  = threadIdx.x;
  const int wave = tid >> 5;
  const int lane = tid & 31;
  const int half = lane >> 4;
  const int l15  = lane & 15;
  const int wm   = wave >> 1;     // 0..3
  const int wn   = wave & 1;      // 0..1
  const int tileM = blockIdx.y * 128;
  const int tileN = blockIdx.x * 64;

  const float* Abase = A + (long long)tileM * lda;
  const float* Bbase = B + (long long)tileN * ldb;

  v8f acc[2][2] = {};

  const int nk = K >> 5;
  if (wave == 0) {
    tdm_load_2d_f32((unsigned)(uintptr_t)&As[0][0][0], Abase, 32, 128, lda, 4, 3);
    tdm_load_2d_f32((unsigned)(uintptr_t)&Bs[0][0][0], Bbase, 32, 64,  ldb, 4, 3);
  }

  for (int t = 0; t < nk; ++t) {
    if (wave == 0) __builtin_amdgcn_s_wait_tensorcnt(0);
    __syncthreads();
    if (wave == 0 && t + 1 < nk) {
      const int nb = (t + 1) & 1;
      const int k0 = (t + 1) << 5;
      tdm_load_2d_f32((unsigned)(uintptr_t)&As[nb][0][0], Abase + k0, 32, 128, lda, 4, 3);
      tdm_load_2d_f32((unsigned)(uintptr_t)&Bs[nb][0][0], Bbase + k0, 32, 64,  ldb, 4, 3);
    }
    const int buf = t & 1;

#pragma unroll
    for (int kk = 0; kk < 32; kk += 4) {
      // A fragments (16x4): M = lane&15, K pair selected by lane half
      v2f a0, a1, b0, b1;
      a0.x = As[buf][wm * 32 + l15][kk + half * 2];
      a0.y = As[buf][wm * 32 + l15][kk + half * 2 + 1];
      a1.x = As[buf][wm * 32 + 16 + l15][kk + half * 2];
      a1.y = As[buf][wm * 32 + 16 + l15][kk + half * 2 + 1];
      // B fragments (4x16): N = lane&15, K pair by lane half (Bs[n][k] = B^T)
      b0.x = Bs[buf][wn * 32 + l15][kk + half * 2];
      b0.y = Bs[buf][wn * 32 + l15][kk + half * 2 + 1];
      b1.x = Bs[buf][wn * 32 + 16 + l15][kk + half * 2];
      b1.y = Bs[buf][wn * 32 + 16 + l15][kk + half * 2 + 1];

      acc[0][0] = wmma_f32(a0, b0, acc[0][0]);
      acc[0][1] = wmma_f32(a0, b1, acc[0][1]);
      acc[1][0] = wmma_f32(a1, b0, acc[1][0]);
      acc[1][1] = wmma_f32(a1, b1, acc[1][1]);
    }
    __syncthreads();   // all reads of buf done before next issue overwrites it
  }

  // store: C fragment row = v + 8*half, col = lane&15
#pragma unroll
  for (int mi = 0; mi < 2; ++mi)
#pragma unroll
    for (int ni = 0; ni < 2; ++ni)
#pragma unroll
      for (int v = 0; v < 8; ++v) {
        int row = tileM + wm * 32 + mi * 16 + 8 * half + v;
        int col = tileN + wn * 32 + ni * 16 + l15;
        C[(long long)row * ldc + col] = acc[mi][ni][v];
      }
}

// =====================================================================
// RMS norm over rows of width 1024 (q latent), in place. 1 block per row.
// =====================================================================
__global__ __launch_bounds__(256)
void rms_rows_1024(float* __restrict__ X, const float* __restrict__ w)
{
  __shared__ float red[8];
  __shared__ float rs_sh;
  const int tid = threadIdx.x;
  const int lane = tid & 31, wave = tid >> 5;
  long long base = (long long)blockIdx.x * 1024;
  float4 xv = *(const float4*)(X + base + tid * 4);
  float ss = xv.x * xv.x + xv.y * xv.y + xv.z * xv.z + xv.w * xv.w;
#pragma unroll
  for (int off = 1; off < 32; off <<= 1) ss += __shfl_xor(ss, off, 32);
  if (lane == 0) red[wave] = ss;
  __syncthreads();
  if (tid == 0) {
    float t = 0.f;
#pragma unroll
    for (int i = 0; i < 8; ++i) t += red[i];
    rs_sh = rsqrtf(t * (1.0f / 1024.0f) + EPSc);
  }
  __syncthreads();
  float rs = rs_sh;
  float4 wv = *(const float4*)(w + tid * 4);
  xv.x *= rs * wv.x; xv.y *= rs * wv.y; xv.z *= rs * wv.z; xv.w *= rs * wv.w;
  *(float4*)(X + base + tid * 4) = xv;
}

// =====================================================================
// RoPE on q_pe (last 64 of each head dim), in place.  1 thread = 1 pair.
// =====================================================================
__global__ __launch_bounds__(256)
void rope_q(float* __restrict__ q, const float* __restrict__ fc)
{
  long long t = (long long)blockIdx.x * 256 + threadIdx.x;  // B*S*NH*32 total
  int p = (int)(t & 31);
  long long row = t >> 5;                    // (b, s, h)
  int s = (int)((row >> 4) & (Sc - 1));      // NH = 16
  float* base = q + row * HDc + NOPEc + 2 * p;
  float x0 = base[0], x1 = base[1];
  float c  = fc[s * 64 + 2 * p];
  float sn = fc[s * 64 + 2 * p + 1];
  base[0] = x0 * c - x1 * sn;
  base[1] = x0 * sn + x1 * c;
}

// =====================================================================
// KV prep: RMS-norm K over HD=128, RoPE on its last 64, scatter K/V into
// per-head caches [B][NKV][KVP][HD] (position s -> slot s+1; slot 0 = sink).
// 1 wave per (b,s,h) row; 4 rows per block.
// =====================================================================
__global__ __launch_bounds__(128)
void kv_prep(const float* __restrict__ kv, const float* __restrict__ w,
             const float* __restrict__ fc,
             float* __restrict__ Kb, float* __restrict__ Vb)
{
  const int wave = threadIdx.x >> 5, lane = threadIdx.x & 31;
  const int row = blockIdx.x * 4 + wave;         // (b*S + s)*NKV + h
  const int h = row & 3;
  const int s = (row >> 2) & (Sc - 1);
  const int b = row >> 13;                       // /(S*NKV)

  const float* src = kv + (long long)row * 256;
  float4 kx = *(const float4*)(src + lane * 4);
  float4 vx = *(const float4*)(src + 128 + lane * 4);

  float ss = kx.x * kx.x + kx.y * kx.y + kx.z * kx.z + kx.w * kx.w;
#pragma unroll
  for (int off = 1; off < 32; off <<= 1) ss += __shfl_xor(ss, off, 32);
  float rs = rsqrtf(ss * (1.0f / 128.0f) + EPSc);

  float4 wv = *(const float4*)(w + lane * 4);
  float4 kn;
  kn.x = kx.x * rs * wv.x; kn.y = kx.y * rs * wv.y;
  kn.z = kx.z * rs * wv.z; kn.w = kx.w * rs * wv.w;

  if (lane >= 16) {   // elements 64..127 -> RoPE pairs p0, p0+1
    int p0 = 2 * lane - 32;
    float c0 = fc[s * 64 + p0 * 2],       s0 = fc[s * 64 + p0 * 2 + 1];
    float c1 = fc[s * 64 + (p0 + 1) * 2], s1 = fc[s * 64 + (p0 + 1) * 2 + 1];
    float o0 = kn.x * c0 - kn.y * s0;
    float o1 = kn.x * s0 + kn.y * c0;
    float o2 = kn.z * c1 - kn.w * s1;
    float o3 = kn.z * s1 + kn.w * c1;
    kn.x = o0; kn.y = o1; kn.z = o2; kn.w = o3;
  }

  long long dst = ((long long)(b * NKVc + h) * KVP + (s + 1)) * HDc + lane * 4;
  *(float4*)(Kb + dst) = kn;
  *(float4*)(Vb + dst) = vx;
}

// sink row (slot 0) + zero padding rows [S+1, KVP)
__global__ __launch_bounds__(128)
void sink_pad(const float* __restrict__ sink,
              float* __restrict__ Kb, float* __restrict__ Vb)
{
  const int bh = blockIdx.x;          // 0..7 ; h = bh & 3
  const int t = threadIdx.x;          // 128
  long long base = (long long)bh * KVP * HDc;
  Kb[base + t] = sink[(bh & 3) * 256 + t];
  Vb[base + t] = sink[(bh & 3) * 256 + 128 + t];
  for (int r = Sc + 1; r < KVP; ++r) {
    Kb[base + (long long)r * HDc + t] = 0.f;
    Vb[base + (long long)r * HDc + t] = 0.f;
  }
}

// =====================================================================
// Flash attention with sink.  grid = (S/64, B*NH), block = 128 (4 waves).
// Each wave owns 16 q rows; Q frags resident; KV streamed 16 rows/iter
// via TDM into double-buffered LDS; scores + P*V via WMMA f32 16x16x4.
// =====================================================================
__global__ __launch_bounds__(128)
void attn_flash(const float* __restrict__ Q, const float* __restrict__ Kb,
                const float* __restrict__ Vb, float* __restrict__ O)
{
  const int bh  = blockIdx.y;
  const int b   = bh >> 4;
  const int h   = bh & 15;
  const int kvh = h >> 2;                    // GQA: 4 q heads per kv head
  const int wave = threadIdx.x >> 5, lane = threadIdx.x & 31;
  const int half = lane >> 4, l15 = lane & 15;
  const int q0 = blockIdx.x * 64 + wave * 16;

  __shared__ float Ks[2][16][132];   // 128 + 4 pad (TDM pad codes 6,3)
  __shared__ float Vs[2][16][132];
  __shared__ float Ps[4][16][20];

  // resident Q fragments: A-layout, M = lane&15 (row q0+l15), K pair by half
  v2f qf[32];
  {
    const float* qrow = Q + (((long long)b * Sc + (q0 + l15)) * NHc + h) * HDc;
#pragma unroll
    for (int kk = 0; kk < 32; ++kk) {
      qf[kk].x = qrow[kk * 4 + 2 * half];
      qf[kk].y = qrow[kk * 4 + 2 * half + 1];
    }
  }

  v8f o[8] = {};
  float m_run[8], l_run[8];
#pragma unroll
  for (int v = 0; v < 8; ++v) { m_run[v] = -3.0e38f; l_run[v] = 0.f; }

  const float* Kbase = Kb + (long long)(b * NKVc + kvh) * KVP * HDc;
  const float* Vbase = Vb + (long long)(b * NKVc + kvh) * KVP * HDc;

  const int ntiles = blockIdx.x * 4 + 5;   // covers kv j <= 64*qtile + 64
  if (wave == 0) {
    tdm_load_2d_f32((unsigned)(uintptr_t)&Ks[0][0][0], Kbase, 128, 16, HDc, 6, 3);
    tdm_load_2d_f32((unsigned)(uintptr_t)&Vs[0][0][0], Vbase, 128, 16, HDc, 6, 3);
  }

  for (int t = 0; t < ntiles; ++t) {
    if (wave == 0) __builtin_amdgcn_s_wait_tensorcnt(0);
    __syncthreads();
    if (wave == 0 && t + 1 < ntiles) {
      const int nb = (t + 1) & 1;
      const long long j0n = (long long)(t + 1) * 16 * HDc;
      tdm_load_2d_f32((unsigned)(uintptr_t)&Ks[nb][0][0], Kbase + j0n, 128, 16, HDc, 6, 3);
      tdm_load_2d_f32((unsigned)(uintptr_t)&Vs[nb][0][0], Vbase + j0n, 128, 16, HDc, 6, 3);
    }
    const int buf = t & 1;
    const int j0 = t * 16;

    // scores: s(16x16) = Q(16x128) * K_tile(16x128)^T
    v8f s = {};
#pragma unroll
    for (int kk = 0; kk < 32; ++kk) {
      v2f bf;
      bf.x = Ks[buf][l15][kk * 4 + 2 * half];
      bf.y = Ks[buf][l15][kk * 4 + 2 * half + 1];
      s = wmma_f32(qf[kk], bf, s);
    }

    // mask + scale + online softmax (row = v + 8*half, col = l15)
    const int kj = j0 + l15;
#pragma unroll
    for (int v = 0; v < 8; ++v) {
      int qi = q0 + 8 * half + v;
      bool vis = (kj == 0) || (kj - 1 <= qi);
      float sv = vis ? s[v] * SCALEc : -3.0e38f;

      float rm = sv;
      rm = fmaxf(rm, __shfl_xor(rm, 1, 32));
      rm = fmaxf(rm, __shfl_xor(rm, 2, 32));
      rm = fmaxf(rm, __shfl_xor(rm, 4, 32));
      rm = fmaxf(rm, __shfl_xor(rm, 8, 32));

      float mn  = fmaxf(m_run[v], rm);
      float fac = __expf(m_run[v] - mn);
      m_run[v] = mn;

      float pv = __expf(sv - mn);
      float rsum = pv;
      rsum += __shfl_xor(rsum, 1, 32);
      rsum += __shfl_xor(rsum, 2, 32);
      rsum += __shfl_xor(rsum, 4, 32);
      rsum += __shfl_xor(rsum, 8, 32);
      l_run[v] = l_run[v] * fac + rsum;

#pragma unroll
      for (int f = 0; f < 8; ++f) o[f][v] *= fac;
      Ps[wave][8 * half + v][l15] = pv;
    }
    __syncthreads();

    // o += P(16x16) * V(16x128)
#pragma unroll
    for (int kk = 0; kk < 4; ++kk) {
      v2f a;
      a.x = Ps[wave][l15][kk * 4 + 2 * half];
      a.y = Ps[wave][l15][kk * 4 + 2 * half + 1];
#pragma unroll
      for (int f = 0; f < 8; ++f) {
        v2f bf;
        bf.x = Vs[buf][kk * 4 + 2 * half][f * 16 + l15];
        bf.y = Vs[buf][kk * 4 + 2 * half + 1][f * 16 + l15];
        o[f] = wmma_f32(a, bf, o[f]);
      }
    }
    // next iteration's top barrier protects buf^1 before its TDM overwrite
  }

  // epilogue: normalize and store (B,S,NH*HD)
#pragma unroll
  for (int v = 0; v < 8; ++v) {
    float inv = 1.0f / l_run[v];
    int qi = q0 + 8 * half + v;
    float* dst = O + (((long long)b * Sc + qi) * NHc + h) * HDc;
#pragma unroll
    for (int f = 0; f < 8; ++f) dst[f * 16 + l15] = o[f][v] * inv;
  }
}

// =====================================================================
extern "C" void kernel_launch(void* const* d_in, const int* in_sizes, int n_in,
                              void* d_out, int out_size, void* d_ws, size_t ws_size,
                              hipStream_t stream) {
  const float* x        = (const float*)d_in[0];
  const float* fc       = (const float*)d_in[1];   // (S, 32, 2)
  const float* wq_a     = (const float*)d_in[2];   // (1024, 2048)
  const float* q_norm_w = (const float*)d_in[3];   // (1024,)
  const float* wq_b     = (const float*)d_in[4];   // (2048, 1024)
  const float* wkv      = (const float*)d_in[5];   // (1024, 2048)
  const float* kv_norm_w= (const float*)d_in[6];   // (128,)
  const float* wo_a     = (const float*)d_in[7];   // (8, 128, 256)
  const float* wo_b     = (const float*)d_in[8];   // (2048, 1024)
  const float* sink     = (const float*)d_in[9];   // (4, 256)
  float* out = (float*)d_out;
  float* ws  = (float*)d_ws;

  const int MT = Bc * Sc;  // 4096 token rows

  // workspace layout (floats); total ~33.6M floats (~134 MB)
  size_t off = 0;
  float* qlat = ws + off; off += (size_t)MT * QLRc;           // 4096x1024
  float* qbuf = ws + off; off += (size_t)MT * NHc * HDc;      // 4096x2048
  float* kvb  = ws + off; off += (size_t)MT * NKVc * 256;     // 4096x1024
  float* Kb   = ws + off; off += (size_t)Bc * NKVc * KVP * HDc;
  float* Vb   = ws + off; off += (size_t)Bc * NKVc * KVP * HDc;
  float* obuf = ws + off; off += (size_t)MT * NHc * HDc;      // 4096x2048
  float* lob  = ws + off; off += (size_t)MT * OLRc;           // 4096x1024

  // 1) q_lat = x @ wq_a^T          [4096 x 1024, K=2048]
  gemm_nt_wmma<<<dim3(QLRc / 64, MT / 128, 1), 256, 0, stream>>>(
      x, wq_a, qlat, MT, QLRc, Hc, Hc, Hc, QLRc, 0, 0, 0);

  // 2) RMS norm rows of q_lat
  rms_rows_1024<<<MT, 256, 0, stream>>>(qlat, q_norm_w);

  // 3) q = q_lat @ wq_b^T          [4096 x 2048, K=1024]
  gemm_nt_wmma<<<dim3((NHc * HDc) / 64, MT / 128, 1), 256, 0, stream>>>(
      qlat, wq_b, qbuf, MT, NHc * HDc, QLRc, QLRc, QLRc, NHc * HDc, 0, 0, 0);

  // 4) kv = x @ wkv^T              [4096 x 1024, K=2048]
  gemm_nt_wmma<<<dim3((NKVc * 256) / 64, MT / 128, 1), 256, 0, stream>>>(
      x, wkv, kvb, MT, NKVc * 256, Hc, Hc, Hc, NKVc * 256, 0, 0, 0);

  // 5) RoPE on q_pe (in place)
  rope_q<<<(Bc * Sc * NHc * 32) / 256, 256, 0, stream>>>(qbuf, fc);

  // 6) KV prep: RMS + RoPE + scatter into caches (slot s+1)
  kv_prep<<<(MT * NKVc) / 4, 128, 0, stream>>>(kvb, kv_norm_w, fc, Kb, Vb);

  // 7) sink rows + padding
  sink_pad<<<Bc * NKVc, 128, 0, stream>>>(sink, Kb, Vb);

  // 8) flash attention
  attn_flash<<<dim3(Sc / 64, Bc * NHc), 128, 0, stream>>>(qbuf, Kb, Vb, obuf);

  // 9) grouped low-rank output proj: lo[.., g] = o_g @ wo_a[g]^T  (8 batches)
  gemm_nt_wmma<<<dim3(128 / 64, MT / 128, OGc), 256, 0, stream>>>(
      obuf, wo_a, lob, MT, 128, 256, NHc * HDc, 256, OLRc,
      /*aBatch=*/256, /*bBatch=*/128 * 256, /*cBatch=*/128);

  // 10) out = lo @ wo_b^T          [4096 x 2048, K=1024]
  gemm_nt_wmma<<<dim3(Hc / 64, MT / 128, 1), 256, 0, stream>>>(
      lob, wo_b, out, MT, Hc, OLRc, OLRc, OLRc, Hc, 0, 0, 0);
}